// AvgPoolVectorsPerWSI_927712936495
// MI455X (gfx1250) — compile-verified
//
#include <hip/hip_runtime.h>

// ---------------------------------------------------------------------------
// AvgPoolVectorsPerWSI for MI455X (gfx1250, wave32)
//
// Roofline: 2048*512*49*4B = 205.5 MB of mandatory reads / 23.3 TB/s ~= 8.8us.
// Phase A (spatial sum) is the bandwidth phase -> float4 (b128) loads.
// Phase B (segment sum) = S^T (64x2048 one-hot) @ X_sp (2048x512), done with
// V_WMMA_F32_16X16X4_F32 (exact f32, one-hot entries are exact). Tiny cost.
// ---------------------------------------------------------------------------

typedef __attribute__((ext_vector_type(2))) float v2f;
typedef __attribute__((ext_vector_type(8))) float v8f;

#define NSAMP  2048   // N
#define MDIM   512    // M (feature dim)
#define NGROUP 64     // segments
#define HW     49     // 7*7

// ---------------------------------------------------------------------------
// Phase A: spatial sum. One thread owns 4 consecutive (n,m) rows:
// 4 rows * 49 floats = 784 bytes = 49 float4, and 784 % 16 == 0, so every
// thread's base is 16B aligned -> pure global_load_b128 traffic.
// The i-th float4 straddles row boundaries at compile-time-known positions;
// full unroll makes the (4i+j)/49 segment indices constants.
// ---------------------------------------------------------------------------
__global__ __launch_bounds__(256) void ap_spatial_sum(const float4* __restrict__ x4,
                                                      float4* __restrict__ xsp4) {
  int tid = blockIdx.x * 256 + threadIdx.x;       // 0 .. 2048*512/4 - 1
  const float4* p = x4 + (size_t)tid * HW;        // 49 float4 = 4 rows
  float acc[4] = {0.f, 0.f, 0.f, 0.f};
#pragma unroll
  for (int i = 0; i < HW; ++i) {
    float4 v = p[i];
    acc[(4 * i + 0) / HW] += v.x;
    acc[(4 * i + 1) / HW] += v.y;
    acc[(4 * i + 2) / HW] += v.z;
    acc[(4 * i + 3) / HW] += v.w;
  }
  xsp4[tid] = make_float4(acc[0], acc[1], acc[2], acc[3]);  // xsp[n][m], aligned
}

// ---------------------------------------------------------------------------
// Counts per group: one block, LDS atomics, then write the 64 ints.
// ---------------------------------------------------------------------------
__global__ __launch_bounds__(1024) void ap_counts(const int* __restrict__ idx,
                                                  int* __restrict__ counts) {
  __shared__ int sc[NGROUP];
  if (threadIdx.x < NGROUP) sc[threadIdx.x] = 0;
  __syncthreads();
  atomicAdd(&sc[idx[threadIdx.x]], 1);
  atomicAdd(&sc[idx[threadIdx.x + 1024]], 1);
  __syncthreads();
  if (threadIdx.x < NGROUP) counts[threadIdx.x] = sc[threadIdx.x];
}

// ---------------------------------------------------------------------------
// Phase B: group_sum = S^T @ X_sp via V_WMMA_F32_16X16X4_F32.
// Output [64 x 512] -> 4 x 32 tiles of 16x16; one wave per tile,
// 8 waves/block, 16 blocks. idx[] staged once in LDS (8 KB) per block.
//
// VGPR layouts (ISA 7.12.2, 32-bit, wave32):
//   A (16x4):  lane&15 = M row, lane>>4 selects K pair, vgpr = K&1
//   B (4x16):  lane&15 = N col, lane>>4 selects K pair, vgpr = K&1
//   C/D:       vgpr v holds M = v + 8*(lane>>4), N = lane&15
// Two accumulators (even/odd K-quad) break the WMMA D->C dependency chain.
// ---------------------------------------------------------------------------
__global__ __launch_bounds__(256) void ap_wmma_segsum(const float* __restrict__ xsp,
                                                      const int* __restrict__ idx,
                                                      const int* __restrict__ counts,
                                                      float* __restrict__ out) {
  __shared__ int sidx[NSAMP];
  for (int i = threadIdx.x; i < NSAMP; i += 256) sidx[i] = idx[i];
  __syncthreads();

  const int wave = threadIdx.x >> 5;
  const int lane = threadIdx.x & 31;
  const int tile = blockIdx.x * 8 + wave;    // 0..127
  const int g_base = (tile & 3) * 16;        // 4 group tiles
  const int m_base = (tile >> 2) * 16;       // 32 feature tiles
  const int half = lane >> 4;                // K-pair selector
  const int ln = lane & 15;                  // M (for A) / N (for B)

  const float* bp = xsp + m_base + ln;       // column base for B loads

  v8f c0 = {};
  v8f c1 = {};

  for (int k0 = 0; k0 < NSAMP; k0 += 8) {
    {
      const int kb = k0 + half * 2;
      v2f a, b;
      a.x = (sidx[kb]     == g_base + ln) ? 1.0f : 0.0f;
      a.y = (sidx[kb + 1] == g_base + ln) ? 1.0f : 0.0f;
      b.x = bp[(size_t)kb * MDIM];
      b.y = bp[(size_t)(kb + 1) * MDIM];
      c0 = __builtin_amdgcn_wmma_f32_16x16x4_f32(false, a, false, b,
                                                 (short)0, c0, false, false);
    }
    {
      const int kb = k0 + 4 + half * 2;
      v2f a, b;
      a.x = (sidx[kb]     == g_base + ln) ? 1.0f : 0.0f;
      a.y = (sidx[kb + 1] == g_base + ln) ? 1.0f : 0.0f;
      b.x = bp[(size_t)kb * MDIM];
      b.y = bp[(size_t)(kb + 1) * MDIM];
      c1 = __builtin_amdgcn_wmma_f32_16x16x4_f32(false, a, false, b,
                                                 (short)0, c1, false, false);
    }
  }

  v8f c = c0 + c1;
#pragma unroll
  for (int v = 0; v < 8; ++v) {
    const int g = g_base + half * 8 + v;
    const int m = m_base + ln;
    const float cnt = (float)counts[g];
    out[g * MDIM + m] = c[v] / (cnt * 49.0f);   // [G, M, 1, 1]
  }
}

// ---------------------------------------------------------------------------
// Launch: in[0] = x (f32, 2048*512*49), in[1] = idx (i32, 2048)
// out = f32 [64, 512, 1, 1].  ws: 4 MB xsp + 256 B counts.
// ---------------------------------------------------------------------------
extern "C" void kernel_launch(void* const* d_in, const int* in_sizes, int n_in,
                              void* d_out, int out_size, void* d_ws, size_t ws_size,
                              hipStream_t stream) {
  const float4* x4  = (const float4*)d_in[0];
  const int*    idx = (const int*)d_in[1];
  float*        out = (float*)d_out;

  float4* xsp4  = (float4*)d_ws;
  float*  xsp   = (float*)d_ws;
  int*    counts = (int*)((char*)d_ws + (size_t)NSAMP * MDIM * sizeof(float));

  // Phase A: 2048*512/4 threads, 4 rows each.
  ap_spatial_sum<<<(NSAMP * MDIM / 4) / 256, 256, 0, stream>>>(x4, xsp4);

  // Counts.
  ap_counts<<<1, 1024, 0, stream>>>(idx, counts);

  // Phase B: 128 tiles, 8 waves/block -> 16 blocks.
  ap_wmma_segsum<<<16, 256, 0, stream>>>(xsp, idx, counts, out);
}